// GATGCNEncoder_22273700397757
// MI455X (gfx1250) — compile-verified
//
#include <hip/hip_runtime.h>
#include <math.h>

#define NN 10000
#define EE 160000
#define ETOT 170000      // EE + NN self loops
#define FIN 4
#define DD 128
#define HH 8
#define HD 1024          // HH * DD
#define NEG_SLOPE 0.2f

typedef float v2f __attribute__((ext_vector_type(2)));
typedef float v8f __attribute__((ext_vector_type(8)));

__device__ __forceinline__ float gelu_exact(float x) {
    return 0.5f * x * (1.0f + erff(x * 0.70710678118654752f));
}

// monotonic float -> uint encoding for atomicMax-based segment max
__device__ __forceinline__ unsigned enc_ord(float f) {
    unsigned u = __float_as_uint(f);
    return (u & 0x80000000u) ? ~u : (u | 0x80000000u);
}
__device__ __forceinline__ float dec_ord(unsigned u) {
    return (u & 0x80000000u) ? __uint_as_float(u & 0x7FFFFFFFu)
                             : __uint_as_float(~u);
}

__global__ void k_zero(float* __restrict__ p, int n) {
    int i = blockIdx.x * blockDim.x + threadIdx.x;
    if (i < n) p[i] = 0.0f;
}

// ---------------- GAT linear: h = x @ gat_w, one fp32 WMMA (K=4) per tile
__global__ void k_gat_linear(const float* __restrict__ x,
                             const float* __restrict__ w,
                             float* __restrict__ h) {
    int wid  = (blockIdx.x * blockDim.x + threadIdx.x) >> 5;   // wave-uniform
    int lane = threadIdx.x & 31;
    if (wid >= (NN / 16) * (HD / 16)) return;
    int mt = wid >> 6;            // 64 column tiles
    int nt = wid & 63;
    int m0 = mt * 16, n0 = nt * 16;
    int lrow = lane & 15;
    int kb   = (lane >> 4) * 2;   // lanes 0-15: K=0,1 ; lanes 16-31: K=2,3
    v2f a = *(const v2f*)(x + (m0 + lrow) * FIN + kb);
    v2f b;
    b.x = w[(kb + 0) * HD + n0 + lrow];
    b.y = w[(kb + 1) * HD + n0 + lrow];
    v8f c = {};
    c = __builtin_amdgcn_wmma_f32_16x16x4_f32(false, a, false, b,
                                              (short)0, c, false, false);
    int hi = lane >> 4;
#pragma unroll
    for (int r = 0; r < 8; ++r)
        h[(m0 + r + 8 * hi) * HD + n0 + lrow] = c[r];
}

// ---------------- per-(node,head) attention scores
__global__ void k_att_scores(const float* __restrict__ h,
                             const float* __restrict__ att_src,
                             const float* __restrict__ att_dst,
                             float* __restrict__ as_, float* __restrict__ ad_) {
    int t = blockIdx.x * blockDim.x + threadIdx.x;
    if (t >= NN * HH) return;
    int n = t >> 3, hd = t & 7;
    const float* hp = h + n * HD + hd * DD;
    const float* sp = att_src + hd * DD;
    const float* dp = att_dst + hd * DD;
    float ss = 0.f, dd = 0.f;
    for (int i = 0; i < DD; ++i) { float v = hp[i]; ss += v * sp[i]; dd += v * dp[i]; }
    as_[t] = ss;
    ad_[t] = dd;
}

// ---------------- edge pass 1: leaky score, segment max, degree
__global__ void k_edge_score(const int* __restrict__ ei,
                             const float* __restrict__ as_,
                             const float* __restrict__ ad_,
                             float* __restrict__ e_out,
                             unsigned* __restrict__ menc,
                             float* __restrict__ deg) {
    int t = blockIdx.x * blockDim.x + threadIdx.x;
    if (t >= ETOT * HH) return;
    int e = t >> 3, hd = t & 7;
    int s, d;
    if (e < EE) { s = ei[e]; d = ei[EE + e]; } else { s = d = e - EE; }
    float v = as_[s * HH + hd] + ad_[d * HH + hd];
    v = v > 0.0f ? v : NEG_SLOPE * v;
    e_out[t] = v;
    atomicMax(&menc[d * HH + hd], enc_ord(v));
    if (hd == 0) atomicAdd(&deg[d], 1.0f);
}

// ---------------- edge pass 2: exp(e - max), segment sum denom
__global__ void k_edge_exp(const int* __restrict__ ei,
                           float* __restrict__ e_inout,
                           const unsigned* __restrict__ menc,
                           float* __restrict__ denom) {
    int t = blockIdx.x * blockDim.x + threadIdx.x;
    if (t >= ETOT * HH) return;
    int e = t >> 3, hd = t & 7;
    int d = (e < EE) ? ei[EE + e] : e - EE;
    float m  = dec_ord(menc[d * HH + hd]);
    float ex = expf(e_inout[t] - m);
    e_inout[t] = ex;
    atomicAdd(&denom[d * HH + hd], ex);
}

// ---------------- GAT aggregate: one wave per edge, 1024 channels
__global__ void k_gat_aggregate(const int* __restrict__ ei,
                                const float* __restrict__ h,
                                const float* __restrict__ exv,
                                const float* __restrict__ denom,
                                float* __restrict__ acc) {
    int wid  = (blockIdx.x * blockDim.x + threadIdx.x) >> 5;
    int lane = threadIdx.x & 31;
    if (wid >= ETOT) return;
    int e = wid, s, d;
    if (e < EE) { s = ei[e]; d = ei[EE + e]; } else { s = d = e - EE; }
    const float* hs = h + s * HD;
    float* ap = acc + d * HD;
#pragma unroll
    for (int i = 0; i < HD / 32; ++i) {       // head index i>>2 is constant per chunk
        int head = i >> 2;
        float al = exv[e * HH + head] / denom[d * HH + head];
        int c = i * 32 + lane;
        atomicAdd(&ap[c], hs[c] * al);
    }
}

// ---------------- elementwise: out = gelu(acc + b[ch])
__global__ void k_bias_gelu(const float* __restrict__ acc,
                            const float* __restrict__ b,
                            float* __restrict__ out, int n, int mask) {
    int i = blockIdx.x * blockDim.x + threadIdx.x;
    if (i >= n) return;
    out[i] = gelu_exact(acc[i] + b[i & mask]);
}

__global__ void k_bias_add(float* __restrict__ out,
                           const float* __restrict__ b, int n, int mask) {
    int i = blockIdx.x * blockDim.x + threadIdx.x;
    if (i >= n) return;
    out[i] = out[i] + b[i & mask];
}

__global__ void k_dinv(const float* __restrict__ deg, float* __restrict__ dinv) {
    int i = blockIdx.x * blockDim.x + threadIdx.x;
    if (i >= NN) return;
    float dg = deg[i];
    dinv[i] = dg > 0.0f ? rsqrtf(dg) : 0.0f;
}

// ---------------- GCN GEMM: Z[NN,128] = X[NN,K] @ W[K,128], fp32 WMMA K-loop
// One wave computes a 16x128 strip: A reused across 8 column tiles.
template <int K>
__global__ void k_gemm_nd(const float* __restrict__ X,
                          const float* __restrict__ W,
                          float* __restrict__ Z) {
    int wid  = (blockIdx.x * blockDim.x + threadIdx.x) >> 5;   // wave-uniform
    int lane = threadIdx.x & 31;
    if (wid >= NN / 16) return;
    int m0   = wid * 16;
    int lrow = lane & 15;
    int kb   = (lane >> 4) * 2;
    v8f zero = {};
    v8f acc[8];
#pragma unroll
    for (int j = 0; j < 8; ++j) acc[j] = zero;
    const float* xrow = X + (m0 + lrow) * K + kb;
    for (int k = 0; k < K; k += 4) {
        v2f a = *(const v2f*)(xrow + k);
#pragma unroll
        for (int j = 0; j < 8; ++j) {
            int col = j * 16 + lrow;
            v2f b;
            b.x = W[(k + kb + 0) * DD + col];
            b.y = W[(k + kb + 1) * DD + col];
            acc[j] = __builtin_amdgcn_wmma_f32_16x16x4_f32(
                false, a, false, b, (short)0, acc[j], false, false);
        }
    }
    int hi = lane >> 4;
#pragma unroll
    for (int j = 0; j < 8; ++j)
#pragma unroll
        for (int r = 0; r < 8; ++r)
            Z[(m0 + r + 8 * hi) * DD + j * 16 + lrow] = acc[j][r];
}

// ---------------- GCN scatter: acc[dst] += z[src] * dinv[src]*dinv[dst]
__global__ void k_gcn_scatter(const int* __restrict__ ei,
                              const float* __restrict__ z,
                              const float* __restrict__ dinv,
                              float* __restrict__ acc) {
    int wid  = (blockIdx.x * blockDim.x + threadIdx.x) >> 5;
    int lane = threadIdx.x & 31;
    if (wid >= ETOT) return;
    int e = wid, s, d;
    if (e < EE) { s = ei[e]; d = ei[EE + e]; } else { s = d = e - EE; }
    float nrm = dinv[s] * dinv[d];
    const float* zs = z + s * DD;
    float* ap = acc + d * DD;
#pragma unroll
    for (int i = 0; i < DD / 32; ++i) {
        int c = i * 32 + lane;
        atomicAdd(&ap[c], zs[c] * nrm);
    }
}

static inline int cdiv(int a, int b) { return (a + b - 1) / b; }

extern "C" void kernel_launch(void* const* d_in, const int* in_sizes, int n_in,
                              void* d_out, int out_size, void* d_ws, size_t ws_size,
                              hipStream_t stream) {
    const float* x       = (const float*)d_in[0];
    const int*   ei      = (const int*)d_in[1];
    const float* gat_w   = (const float*)d_in[2];
    const float* att_src = (const float*)d_in[3];
    const float* att_dst = (const float*)d_in[4];
    const float* gat_b   = (const float*)d_in[5];
    const float* gcn1_w  = (const float*)d_in[6];
    const float* gcn1_b  = (const float*)d_in[7];
    const float* gcn2_w  = (const float*)d_in[8];
    const float* gcn2_b  = (const float*)d_in[9];
    float* out = (float*)d_out;
    float* ws  = (float*)d_ws;

    // workspace layout (floats)
    float*    h    = ws;                  // [NN, HD]
    float*    acc  = h    + NN * HD;      // [NN, HD]  GAT accumulation
    float*    as_  = acc  + NN * HD;      // [NN, HH]
    float*    ad_  = as_  + NN * HH;      // [NN, HH]
    unsigned* menc = (unsigned*)(ad_ + NN * HH);  // [NN, HH]
    float*    den  = (float*)menc + NN * HH;      // [NN, HH]
    float*    ev   = den  + NN * HH;      // [ETOT, HH] scores then exp
    float*    deg  = ev   + ETOT * HH;    // [NN]
    float*    dinv = deg  + NN;           // [NN]
    float*    z1   = dinv + NN;           // [NN, DD]
    float*    acc2 = z1   + NN * DD;      // [NN, DD]
    float*    h2   = acc2 + NN * DD;      // [NN, DD]
    float*    z2   = h2   + NN * DD;      // [NN, DD]

    const int B = 256;

    // deterministic re-init of all accumulators every call
    k_zero<<<cdiv(NN * HD, B), B, 0, stream>>>(acc, NN * HD);
    k_zero<<<cdiv(2 * NN * HH, B), B, 0, stream>>>((float*)menc, 2 * NN * HH); // menc+den
    k_zero<<<cdiv(NN, B), B, 0, stream>>>(deg, NN);
    k_zero<<<cdiv(NN * DD, B), B, 0, stream>>>(acc2, NN * DD);
    k_zero<<<cdiv(NN * DD, B), B, 0, stream>>>(out, NN * DD);

    // GAT linear (fp32 WMMA, K=4 single shot): 40000 waves
    k_gat_linear<<<cdiv((NN / 16) * (HD / 16) * 32, B), B, 0, stream>>>(x, gat_w, h);

    // attention scores
    k_att_scores<<<cdiv(NN * HH, B), B, 0, stream>>>(h, att_src, att_dst, as_, ad_);

    // edge passes
    k_edge_score<<<cdiv(ETOT * HH, B), B, 0, stream>>>(ei, as_, ad_, ev, menc, deg);
    k_edge_exp<<<cdiv(ETOT * HH, B), B, 0, stream>>>(ei, ev, menc, den);

    // softmax-weighted message aggregation (one wave per edge)
    k_gat_aggregate<<<cdiv(ETOT * 32, B), B, 0, stream>>>(ei, h, ev, den, acc);

    // bias + exact GELU -> h1 (reuse h buffer)
    k_bias_gelu<<<cdiv(NN * HD, B), B, 0, stream>>>(acc, gat_b, h, NN * HD, HD - 1);

    // symmetric normalization
    k_dinv<<<cdiv(NN, B), B, 0, stream>>>(deg, dinv);

    // GCN layer 1: GEMM (K=1024) then scatter then bias+GELU
    k_gemm_nd<HD><<<cdiv((NN / 16) * 32, B), B, 0, stream>>>(h, gcn1_w, z1);
    k_gcn_scatter<<<cdiv(ETOT * 32, B), B, 0, stream>>>(ei, z1, dinv, acc2);
    k_bias_gelu<<<cdiv(NN * DD, B), B, 0, stream>>>(acc2, gcn1_b, h2, NN * DD, DD - 1);

    // GCN layer 2: GEMM (K=128) then scatter into d_out, bias
    k_gemm_nd<DD><<<cdiv((NN / 16) * 32, B), B, 0, stream>>>(h2, gcn2_w, z2);
    k_gcn_scatter<<<cdiv(ETOT * 32, B), B, 0, stream>>>(ei, z2, dinv, out);
    k_bias_add<<<cdiv(NN * DD, B), B, 0, stream>>>(out, gcn2_b, NN * DD, DD - 1);

    (void)in_sizes; (void)n_in; (void)out_size; (void)ws_size;
}